// GATModel_68401649156180
// MI455X (gfx1250) — compile-verified
//
#include <hip/hip_runtime.h>
#include <hip/hip_bf16.h>

typedef float v2f __attribute__((ext_vector_type(2)));
typedef float v8f __attribute__((ext_vector_type(8)));

#define HEADS 4
#define HID 32
#define HC 128          // HEADS*HID
#define NEG_SLOPE 0.2f
#define NUM_GRAPHS 64

// ---------------- order-preserving float <-> uint for atomicMax ----------------
__device__ __forceinline__ unsigned ordf(float f) {
  unsigned u = __float_as_uint(f);
  return (u & 0x80000000u) ? ~u : (u | 0x80000000u);
}
__device__ __forceinline__ float unordf(unsigned u) {
  u = (u & 0x80000000u) ? (u & 0x7FFFFFFFu) : ~u;
  return __uint_as_float(u);
}

// ---------------- fp32 WMMA GEMM: Y[rows,128] = X[rows,128] @ W[128,128] -------
// One block = 8 waves; each wave owns one 16-wide N tile of a 16-row M tile.
// V_WMMA_F32_16X16X4_F32, 32 WMMAs over K=128.
// W staged in LDS in k-pair layout: Wl[k>>1][n][k&1] so each lane's B fragment
// (two consecutive K values at one column) is a single ds_load_b64.
// A row index is clamped instead of predicated: A-row r only feeds D-row r,
// and out-of-range D rows are never stored, so unconditional b64 loads are safe.
__global__ void gat_gemm128(const float* __restrict__ X,
                            const float* __restrict__ W,
                            float* __restrict__ Y, int rows) {
  __shared__ float Wl[128 * 128];     // 64 KB, k-pair-transposed layout
  for (int i = threadIdx.x; i < 64 * 128; i += blockDim.x) {
    int kp = i >> 7, n = i & 127;
    Wl[kp * 256 + n * 2 + 0] = W[(2 * kp + 0) * 128 + n];
    Wl[kp * 256 + n * 2 + 1] = W[(2 * kp + 1) * 128 + n];
  }
  __syncthreads();

  const int wave = threadIdx.x >> 5;          // 0..7 -> N tile
  const int lane = threadIdx.x & 31;
  const int half = lane >> 4;                 // K/M-row half select
  const int idx  = lane & 15;
  const int mBase = blockIdx.x * 16;
  const int nBase = wave * 16;
  int mrow = mBase + idx;
  if (mrow >= rows) mrow = rows - 1;          // clamp; that D row is never stored
  const float* Xrow = X + (size_t)mrow * 128;
  const float* Bp = Wl + (nBase + idx) * 2 + half * 256;  // + (k>>1)*256 per step

  v8f acc = {};
#pragma unroll
  for (int k = 0; k < 128; k += 4) {
    v2f a = *(const v2f*)(Xrow + k + 2 * half);     // A 16x4: K = k+2h .. k+2h+1
    v2f b = *(const v2f*)(Bp + (k >> 1) * 256);     // B 4x16: same K pair, col idx
    acc = __builtin_amdgcn_wmma_f32_16x16x4_f32(
        /*neg_a=*/false, a, /*neg_b=*/false, b,
        /*c_mod=*/(short)0, acc, /*reuse_a=*/false, /*reuse_b=*/false);
  }
#pragma unroll
  for (int j = 0; j < 8; ++j) {
    int r = mBase + j + 8 * half;               // C/D: VGPR j -> row j + 8*half
    if (r < rows) Y[(size_t)r * 128 + nBase + idx] = acc[j];
  }
}

// ---------------- per-node attention terms: a_s/a_d [N,H] ----------------------
__global__ void gat_node_att(const float* __restrict__ h,
                             const float* __restrict__ att_s,
                             const float* __restrict__ att_d,
                             float* __restrict__ a_s, float* __restrict__ a_d,
                             int nN) {
  int gid = blockIdx.x * blockDim.x + threadIdx.x;
  int n = gid >> 2, head = gid & 3;
  if (n >= nN) return;
  const float* hp = h + (size_t)n * HC + head * HID;
  const float* as = att_s + head * HID;
  const float* ad = att_d + head * HID;
  float s = 0.f, d = 0.f;
#pragma unroll
  for (int c = 0; c < HID; ++c) { float v = hp[c]; s += v * as[c]; d += v * ad[c]; }
  a_s[n * HEADS + head] = s;
  a_d[n * HEADS + head] = d;
}

__device__ __forceinline__ void edge_nodes(const int* __restrict__ ei, int nE,
                                           int edge, int& s, int& d) {
  if (edge < nE) { s = ei[edge]; d = ei[nE + edge]; }
  else           { s = edge - nE; d = edge - nE; }   // self-loops
}

// ---------------- segment max of leaky(e) over dst ----------------------------
__global__ void gat_edge_max(const int* __restrict__ ei, int nE, int nTot,
                             const float* __restrict__ a_s,
                             const float* __restrict__ a_d,
                             unsigned* __restrict__ mord) {
  int gid = blockIdx.x * blockDim.x + threadIdx.x;
  int edge = gid >> 2, head = gid & 3;
  if (edge >= nTot) return;
  int s, d; edge_nodes(ei, nE, edge, s, d);
  float e = a_s[s * HEADS + head] + a_d[d * HEADS + head];
  e = e > 0.f ? e : NEG_SLOPE * e;
  atomicMax(&mord[d * HEADS + head], ordf(e));
}

// ---------------- segment sum of exp(e - m) over dst --------------------------
__global__ void gat_edge_denom(const int* __restrict__ ei, int nE, int nTot,
                               const float* __restrict__ a_s,
                               const float* __restrict__ a_d,
                               const unsigned* __restrict__ mord,
                               float* __restrict__ denom) {
  int gid = blockIdx.x * blockDim.x + threadIdx.x;
  int edge = gid >> 2, head = gid & 3;
  if (edge >= nTot) return;
  int s, d; edge_nodes(ei, nE, edge, s, d);
  float e = a_s[s * HEADS + head] + a_d[d * HEADS + head];
  e = e > 0.f ? e : NEG_SLOPE * e;
  float m = unordf(mord[d * HEADS + head]);
  atomicAdd(&denom[d * HEADS + head], __expf(e - m));
}

// ---------------- weighted scatter: out[dst] += alpha * h[src] ----------------
// One wave32 per edge; each lane handles 4 contiguous channels (same head).
__global__ void gat_edge_scatter(const int* __restrict__ ei, int nE, int nTot,
                                 const float* __restrict__ h,
                                 const float* __restrict__ a_s,
                                 const float* __restrict__ a_d,
                                 const unsigned* __restrict__ mord,
                                 const float* __restrict__ denom,
                                 float* __restrict__ out) {
  int gid = blockIdx.x * blockDim.x + threadIdx.x;
  int edge = gid >> 5, lane = gid & 31;
  if (edge >= nTot) return;
  int s, d; edge_nodes(ei, nE, edge, s, d);
  int c = lane * 4;
  int head = c >> 5;                 // 4 channels always inside one head
  float e = a_s[s * HEADS + head] + a_d[d * HEADS + head];
  e = e > 0.f ? e : NEG_SLOPE * e;
  float m = unordf(mord[d * HEADS + head]);
  float alpha = __expf(e - m) / denom[d * HEADS + head];
  float4 hv = *(const float4*)(h + (size_t)s * HC + c);
  float* o = out + (size_t)d * HC + c;
  atomicAdd(o + 0, hv.x * alpha);
  atomicAdd(o + 1, hv.y * alpha);
  atomicAdd(o + 2, hv.z * alpha);
  atomicAdd(o + 3, hv.w * alpha);
}

// ---------------- bias + relu, in place ---------------------------------------
__global__ void gat_bias_relu(float* __restrict__ buf,
                              const float* __restrict__ bias, int total) {
  int gid = blockIdx.x * blockDim.x + threadIdx.x;
  if (gid >= total) return;
  float v = buf[gid] + bias[gid & (HC - 1)];
  buf[gid] = v > 0.f ? v : 0.f;
}

// ---------------- global mean pool: accumulate sums + counts ------------------
__global__ void gat_pool(const float* __restrict__ buf,
                         const int* __restrict__ batch,
                         float* __restrict__ pooled, float* __restrict__ cnt,
                         int nN) {
  int gid = blockIdx.x * blockDim.x + threadIdx.x;
  int n = gid >> 7, c = gid & (HC - 1);
  if (n >= nN) return;
  int g = batch[n];
  atomicAdd(&pooled[g * HC + c], buf[(size_t)n * HC + c]);
  if (c == 0) atomicAdd(&cnt[g], 1.0f);
}

// ---------------- output heads ------------------------------------------------
__global__ void gat_heads(const float* __restrict__ pooled,
                          const float* __restrict__ cnt,
                          const float* __restrict__ Wr, const float* __restrict__ br,
                          const float* __restrict__ Wt, const float* __restrict__ bt,
                          float* __restrict__ out) {
  int g = blockIdx.x * blockDim.x + threadIdx.x;
  if (g >= NUM_GRAPHS) return;
  float inv = 1.0f / fmaxf(cnt[g], 1.0f);
  float r = 0.f, t = 0.f;
#pragma unroll 4
  for (int c = 0; c < HC; ++c) {
    float p = pooled[g * HC + c] * inv;
    r += p * Wr[c];
    t += p * Wt[c];
  }
  out[g * 2 + 0] = r + br[0];
  out[g * 2 + 1] = t + bt[0];
}

extern "C" void kernel_launch(void* const* d_in, const int* in_sizes, int n_in,
                              void* d_out, int out_size, void* d_ws, size_t ws_size,
                              hipStream_t stream) {
  const float* x     = (const float*)d_in[0];
  const int*   ei    = (const int*)d_in[1];
  const int*   batch = (const int*)d_in[2];
  const float* W1    = (const float*)d_in[3];
  const float* as1   = (const float*)d_in[4];
  const float* ad1   = (const float*)d_in[5];
  const float* b1    = (const float*)d_in[6];
  const float* W2    = (const float*)d_in[7];
  const float* as2   = (const float*)d_in[8];
  const float* ad2   = (const float*)d_in[9];
  const float* b2    = (const float*)d_in[10];
  const float* Wr    = (const float*)d_in[11];
  const float* br    = (const float*)d_in[12];
  const float* Wt    = (const float*)d_in[13];
  const float* bt    = (const float*)d_in[14];
  float* out = (float*)d_out;

  const int N  = in_sizes[0] / HC;   // 40000
  const int E  = in_sizes[1] / 2;    // 640000
  const int ET = E + N;              // edges incl. self-loops

  // workspace carve-up
  float*    bufA   = (float*)d_ws;              // [N, 128]  h = X@W
  float*    bufB   = bufA + (size_t)N * HC;     // [N, 128]  aggregation target
  float*    a_s    = bufB + (size_t)N * HC;     // [N, 4]
  float*    a_d    = a_s + (size_t)N * HEADS;   // [N, 4]
  unsigned* mord   = (unsigned*)(a_d + (size_t)N * HEADS);  // [N, 4]
  float*    denom  = (float*)(mord + (size_t)N * HEADS);    // [N, 4]
  float*    pooled = denom + (size_t)N * HEADS;             // [64, 128]
  float*    cnt    = pooled + NUM_GRAPHS * HC;              // [64]

  const int TB = 256;
  dim3 blk(TB);
  const int gGemm    = (N + 15) / 16;
  const int gNode    = (N * HEADS + TB - 1) / TB;
  const int gEdge4   = (ET * HEADS + TB - 1) / TB;
  const int gEdge32  = ((ET * 32) + TB - 1) / TB;
  const int gElem    = (N * HC + TB - 1) / TB;

  // ================= layer 1 =================
  gat_gemm128<<<gGemm, blk, 0, stream>>>(x, W1, bufA, N);
  gat_node_att<<<gNode, blk, 0, stream>>>(bufA, as1, ad1, a_s, a_d, N);
  (void)hipMemsetAsync(bufB, 0, (size_t)N * HC * sizeof(float), stream);
  (void)hipMemsetAsync(mord, 0, (size_t)N * HEADS * sizeof(unsigned), stream);
  (void)hipMemsetAsync(denom, 0, (size_t)N * HEADS * sizeof(float), stream);
  gat_edge_max<<<gEdge4, blk, 0, stream>>>(ei, E, ET, a_s, a_d, mord);
  gat_edge_denom<<<gEdge4, blk, 0, stream>>>(ei, E, ET, a_s, a_d, mord, denom);
  gat_edge_scatter<<<gEdge32, blk, 0, stream>>>(ei, E, ET, bufA, a_s, a_d, mord, denom, bufB);
  gat_bias_relu<<<gElem, blk, 0, stream>>>(bufB, b1, N * HC);

  // ================= layer 2 =================
  gat_gemm128<<<gGemm, blk, 0, stream>>>(bufB, W2, bufA, N);
  gat_node_att<<<gNode, blk, 0, stream>>>(bufA, as2, ad2, a_s, a_d, N);
  (void)hipMemsetAsync(bufB, 0, (size_t)N * HC * sizeof(float), stream);
  (void)hipMemsetAsync(mord, 0, (size_t)N * HEADS * sizeof(unsigned), stream);
  (void)hipMemsetAsync(denom, 0, (size_t)N * HEADS * sizeof(float), stream);
  gat_edge_max<<<gEdge4, blk, 0, stream>>>(ei, E, ET, a_s, a_d, mord);
  gat_edge_denom<<<gEdge4, blk, 0, stream>>>(ei, E, ET, a_s, a_d, mord, denom);
  gat_edge_scatter<<<gEdge32, blk, 0, stream>>>(ei, E, ET, bufA, a_s, a_d, mord, denom, bufB);
  gat_bias_relu<<<gElem, blk, 0, stream>>>(bufB, b2, N * HC);

  // ================= pool + heads =================
  (void)hipMemsetAsync(pooled, 0, NUM_GRAPHS * HC * sizeof(float), stream);
  (void)hipMemsetAsync(cnt, 0, NUM_GRAPHS * sizeof(float), stream);
  gat_pool<<<gElem, blk, 0, stream>>>(bufB, batch, pooled, cnt, N);
  gat_heads<<<1, 64, 0, stream>>>(pooled, cnt, Wr, br, Wt, bt, out);
}